// BaldNeRF_89318139888247
// MI455X (gfx1250) — compile-verified
//
#include <hip/hip_runtime.h>
#include <math.h>

// ---------------------------------------------------------------------------
// BaldNeRF density sampling for MI455X (gfx1250, wave32).
//
// Stage 1 (colstats_kernel): per-(h,w)-column softmax statistics over depth.
//   colmax[c]   = max_d fv[d,c]
//   colscale[c] = (relu(lmax[c]) + 0.01) / sum_d exp(fv[d,c] - colmax[c])
//   => density(d,c) = exp(fv[d,c] - colmax[c]) * colscale[c]
//   Avoids materializing the 78.6 MB density volume; the raw volume stays hot
//   in the 192 MB L2 for the gather stage.
//
// Stage 2 (sample_kernel): camera projection via V_WMMA_F32_16X16X4_F32 (the
//   reference's einsum with the 3x3 intrinsics), then border-clamped,
//   align_corners trilinear sampling with softmax applied on the fly at the
//   8 gathered voxels. Coordinate math is fused algebraically and uses
//   v_rcp_f32 instead of IEEE divide expansions.
// ---------------------------------------------------------------------------

typedef float v2f __attribute__((ext_vector_type(2)));
typedef float v8f __attribute__((ext_vector_type(8)));

#define EPS_F 1e-10f

// Volume geometry from the reference: feature_volume [1,1,256,240,320]
constexpr int VD  = 256;
constexpr int VH  = 240;
constexpr int VW  = 320;
constexpr int VHW = VH * VW;  // 76800 columns

// ---------------------------------------------------------------------------
// Stage 1: column softmax stats. One thread per (h,w) column; at each depth
// step a wave reads 32 consecutive w's -> fully coalesced streaming.
// ---------------------------------------------------------------------------
__global__ __launch_bounds__(256) void colstats_kernel(
    const float* __restrict__ fv,
    const float* __restrict__ lmax,
    float* __restrict__ colmax,
    float* __restrict__ colscale)
{
    int col = blockIdx.x * blockDim.x + threadIdx.x;
    if (col >= VHW) return;
    const float* p = fv + col;

    float m = -INFINITY;
    #pragma unroll 4
    for (int d = 0; d < VD; ++d)
        m = fmaxf(m, p[d * VHW]);

    float s = 0.f;
    #pragma unroll 4
    for (int d = 0; d < VD; ++d)
        s += __expf(p[d * VHW] - m);

    float sc = lmax[col];
    sc = (sc > 0.f ? sc : 0.f) + 0.01f;   // relu(max) + 0.01

    colmax[col]   = m;
    colscale[col] = sc * __builtin_amdgcn_rcpf(s);
}

// ---------------------------------------------------------------------------
// Stage 2: projection (WMMA) + trilinear density sampling.
//
// WMMA mapping (D = A x B, M=16,N=16,K=4, fp32):
//   A (16x4)  = camera intrinsics: row j (j<3) = intr[j][0..2], K=3 padded 0.
//               Per ISA layout: lanes 0-15 hold {K0,K1}, lanes 16-31 {K2,K3}.
//   B (4x16)  = 16 points as columns: K0=x, K1=y, K2=z, K3=0.
//               z rows live in lanes 16-31 -> shfl_xor(...,16) partner data.
//   D         : VGPR r, lanes 0-15 => M=r, N=lane -> D[0]=px, D[1]=py,
//               D[2]=pz. Two WMMAs cover a wave's 32 points; a final
//               shfl_xor(...,16) redistributes WMMA#2's results to lanes
//               16-31.
// ---------------------------------------------------------------------------
__global__ __launch_bounds__(256) void sample_kernel(
    const float* __restrict__ fv,
    const float* __restrict__ colmax,
    const float* __restrict__ colscale,
    const float* __restrict__ intr,
    const float* __restrict__ pts,
    const int* __restrict__ Hp, const int* __restrict__ Wp,
    const int* __restrict__ dminp, const int* __restrict__ dmaxp,
    float* __restrict__ out, int P)
{
    int t  = blockIdx.x * 256 + threadIdx.x;
    int tc = (t < P) ? t : (P - 1);        // clamp loads; keep EXEC all-1s

    float x = pts[3 * tc + 0];
    float y = pts[3 * tc + 1];
    float z = pts[3 * tc + 2];

    // ---- projection via V_WMMA_F32_16X16X4_F32 ----
    unsigned lane = threadIdx.x & 31u;
    bool     hi   = lane >= 16u;
    unsigned mrow = lane & 15u;

    float a0 = 0.f, a1 = 0.f;
    if (mrow < 3u) {
        if (!hi) { a0 = intr[mrow * 3 + 0]; a1 = intr[mrow * 3 + 1]; }
        else     { a0 = intr[mrow * 3 + 2]; /* K=3 column stays 0 */ }
    }
    v2f A = {a0, a1};

    float xo = __shfl_xor(x, 16, 32);
    float yo = __shfl_xor(y, 16, 32);
    float zo = __shfl_xor(z, 16, 32);

    v2f B1 = { hi ? zo : x, hi ? 0.f : y };   // points of lanes 0..15
    v2f B2 = { hi ? z : xo, hi ? 0.f : yo };  // points of lanes 16..31
    v8f Cz = {0.f, 0.f, 0.f, 0.f, 0.f, 0.f, 0.f, 0.f};

    v8f D1 = __builtin_amdgcn_wmma_f32_16x16x4_f32(
        false, A, false, B1, (short)0, Cz, false, false);
    v8f D2 = __builtin_amdgcn_wmma_f32_16x16x4_f32(
        false, A, false, B2, (short)0, Cz, false, false);

    float px2 = __shfl_xor(D2[0], 16, 32);
    float py2 = __shfl_xor(D2[1], 16, 32);
    float pz2 = __shfl_xor(D2[2], 16, 32);

    float px = hi ? px2 : D1[0];
    float py = hi ? py2 : D1[1];
    float pz = hi ? pz2 : D1[2];

    // ---- fused grid coords -> voxel coords ----
    // fx = (gx+1)*0.5*(VW-1) with gx = (px/(pz+EPS)/W - 0.5)*2
    //    = px * invpz * ((VW-1)/W)            (and likewise fy)
    // fz = (1/pz - dmin) * ((VD-1)/(dmax-dmin))
    float Wf   = (float)(*Wp);
    float Hf   = (float)(*Hp);
    float dmin = (float)(*dminp);
    float dmax = (float)(*dmaxp);

    float sx = (float)(VW - 1) * __builtin_amdgcn_rcpf(Wf);
    float sy = (float)(VH - 1) * __builtin_amdgcn_rcpf(Hf);
    float sz = (float)(VD - 1) * __builtin_amdgcn_rcpf(dmax - dmin);

    float invpz = __builtin_amdgcn_rcpf(pz + EPS_F);
    float rz    = __builtin_amdgcn_rcpf(pz);       // reference: no EPS for gz

    float fx = fminf(fmaxf(px * invpz * sx, 0.f), (float)(VW - 1));
    float fy = fminf(fmaxf(py * invpz * sy, 0.f), (float)(VH - 1));
    float fz = fminf(fmaxf((rz - dmin) * sz, 0.f), (float)(VD - 1));

    int x0 = (int)floorf(fx); int x1 = min(x0 + 1, VW - 1);
    int y0 = (int)floorf(fy); int y1 = min(y0 + 1, VH - 1);
    int z0 = (int)floorf(fz); int z1 = min(z0 + 1, VD - 1);
    float wx = fx - (float)x0;
    float wy = fy - (float)y0;
    float wz = fz - (float)z0;

    int c00 = y0 * VW + x0;   // (y0,x0)
    int c01 = y0 * VW + x1;   // (y0,x1)
    int c10 = y1 * VW + x0;   // (y1,x0)
    int c11 = y1 * VW + x1;   // (y1,x1)

    float m00 = colmax[c00], s00 = colscale[c00];
    float m01 = colmax[c01], s01 = colscale[c01];
    float m10 = colmax[c10], s10 = colscale[c10];
    float m11 = colmax[c11], s11 = colscale[c11];

    int off0 = z0 * VHW;
    int off1 = z1 * VHW;

    float v000 = __expf(fv[off0 + c00] - m00) * s00;
    float v001 = __expf(fv[off0 + c01] - m01) * s01;
    float v010 = __expf(fv[off0 + c10] - m10) * s10;
    float v011 = __expf(fv[off0 + c11] - m11) * s11;
    float v100 = __expf(fv[off1 + c00] - m00) * s00;
    float v101 = __expf(fv[off1 + c01] - m01) * s01;
    float v110 = __expf(fv[off1 + c10] - m10) * s10;
    float v111 = __expf(fv[off1 + c11] - m11) * s11;

    float e00 = v000 + (v001 - v000) * wx;  // (z0,y0)
    float e01 = v010 + (v011 - v010) * wx;  // (z0,y1)
    float e10 = v100 + (v101 - v100) * wx;  // (z1,y0)
    float e11 = v110 + (v111 - v110) * wx;  // (z1,y1)
    float e0  = e00 + (e01 - e00) * wy;
    float e1  = e10 + (e11 - e10) * wy;
    float res = e0 + (e1 - e0) * wz;

    if (t < P) out[t] = res;
}

// ---------------------------------------------------------------------------
// Launch. Inputs (setup_inputs order):
//   0: feature_volume f32 [1,1,256,240,320]   1: learnable_max_value f32 [1,1,1,240,320]
//   2: intr f32 [1,3,3]                       3: points f32 [1,8192,128,3]
//   4: H (int) 5: W (int) 6: depth_min (int) 7: depth_max (int)
// Output: f32 [1,8192,128]. Workspace: 2*VHW floats (614 KB).
// ---------------------------------------------------------------------------
extern "C" void kernel_launch(void* const* d_in, const int* in_sizes, int n_in,
                              void* d_out, int out_size, void* d_ws, size_t ws_size,
                              hipStream_t stream)
{
    const float* fv    = (const float*)d_in[0];
    const float* lmax  = (const float*)d_in[1];
    const float* intr  = (const float*)d_in[2];
    const float* pts   = (const float*)d_in[3];
    const int*   Hp    = (const int*)d_in[4];
    const int*   Wp    = (const int*)d_in[5];
    const int*   dminp = (const int*)d_in[6];
    const int*   dmaxp = (const int*)d_in[7];
    float*       outp  = (float*)d_out;

    float* colmax   = (float*)d_ws;
    float* colscale = colmax + VHW;

    colstats_kernel<<<dim3((VHW + 255) / 256), dim3(256), 0, stream>>>(
        fv, lmax, colmax, colscale);

    int P = in_sizes[3] / 3;  // 1,048,576 points (multiple of 256)
    sample_kernel<<<dim3((P + 255) / 256), dim3(256), 0, stream>>>(
        fv, colmax, colscale, intr, pts, Hp, Wp, dminp, dmaxp, outp, P);
}